// QualityAwarePatchAugment_9861244912148
// MI455X (gfx1250) — compile-verified
//
#include <hip/hip_runtime.h>

// Problem geometry (fixed by the reference)
#define PSZ 16
#define BB  32
#define HH  1024
#define WW  1024
#define PH  64
#define PW  64
#define NP  (PH * PW)       // 4096 patches
#define NBP (BB * NP)       // 131072 (batch, patch) pairs

// ---- gfx1250 async LDS staging (ASYNCcnt path), guarded ----
#if defined(__gfx1250__) && __has_builtin(__builtin_amdgcn_global_load_async_to_lds_b32)
  #define USE_ASYNC_LDS 1
  #if __has_builtin(__builtin_amdgcn_s_wait_asynccnt)
    #define WAIT_ASYNC() __builtin_amdgcn_s_wait_asynccnt(0)
  #else
    #define WAIT_ASYNC() asm volatile("s_wait_asynccnt 0" ::: "memory")
  #endif
#else
  #define USE_ASYNC_LDS 0
#endif

#if USE_ASYNC_LDS && __has_builtin(__builtin_amdgcn_global_load_async_to_lds_b128)
  #define USE_ASYNC_LDS128 1
#else
  #define USE_ASYNC_LDS128 0
#endif

typedef int v4i __attribute__((vector_size(16)));   // b128 payload type

__device__ __forceinline__ float clamp01(float v) {
    return fminf(fmaxf(v, 0.0f), 1.0f);
}

// ---------------- Kernel 1: per (b, patch) stats, one wave32 per patch ----------------
// Each lane does two b128 loads (one 16B row chunk from the top half, one from the
// bottom half of the patch) -> 2 vmem instructions per lane instead of 8.
__global__ __launch_bounds__(256) void k_patch_stats(const float* __restrict__ img,
                                                     float* __restrict__ qual_b,
                                                     float* __restrict__ mean_b) {
    const int wave = blockIdx.x * 8 + (threadIdx.x >> 5);   // (b*NP + p)
    const int lane = threadIdx.x & 31;
    const int b = wave >> 12;          // / 4096
    const int p = wave & (NP - 1);
    const int ph = p >> 6, pw = p & 63;
    const float* base = img + (size_t)b * HH * WW + (size_t)ph * PSZ * WW + (size_t)pw * PSZ;

    const int r0 = lane >> 2;                 // row 0..7
    const int c4 = (lane & 3) << 2;           // col 0,4,8,12
    const float* rowp = base + (size_t)r0 * WW + c4;
    const float4 a = *reinterpret_cast<const float4*>(rowp);             // rows 0..7
    const float4 d = *reinterpret_cast<const float4*>(rowp + 8 * WW);    // rows 8..15

    float s1 = a.x + a.y + a.z + a.w + d.x + d.y + d.z + d.w;
    float s2 = a.x * a.x + a.y * a.y + a.z * a.z + a.w * a.w
             + d.x * d.x + d.y * d.y + d.z * d.z + d.w * d.w;

    // wave32 butterfly reduction
#pragma unroll
    for (int m = 16; m >= 1; m >>= 1) {
        s1 += __shfl_xor(s1, m, 32);
        s2 += __shfl_xor(s2, m, 32);
    }
    if (lane == 0) {
        const float mean = s1 * (1.0f / 256.0f);
        float var = (s2 - 256.0f * mean * mean) * (1.0f / 255.0f);   // unbiased (ddof=1)
        var = fmaxf(var, 0.0f);
        const float sd = sqrtf(var);
        const float iq = 1.0f - 2.0f * fabsf(mean - 0.5f);
        qual_b[wave] = (sd + iq + var) * (1.0f / 3.0f);
        mean_b[wave] = mean;
    }
}

// ---------------- Kernel 2: batch-average + per-patch action code ----------------
__global__ __launch_bounds__(256) void k_code(const float* __restrict__ qual_b,
                                              const float* __restrict__ mean_b,
                                              const float* __restrict__ r_strong,
                                              const float* __restrict__ r_drop,
                                              const float* __restrict__ r_else,
                                              const int* __restrict__ aug_choice,
                                              const int* __restrict__ slight_choice,
                                              int* __restrict__ code,
                                              float* __restrict__ mpp) {
    const int p = blockIdx.x * blockDim.x + threadIdx.x;
    if (p >= NP) return;
    float sq = 0.0f, sm = 0.0f;
#pragma unroll
    for (int b = 0; b < BB; ++b) {
        sq += qual_b[b * NP + p];
        sm += mean_b[b * NP + p];
    }
    const float q = sq * (1.0f / (float)BB);
    const float m = sm * (1.0f / (float)BB);

    const bool low    = q < 0.7f;
    const bool strong = low && (r_strong[p] < 0.8f);
    const bool drop   = low && (q < 0.3f) && (r_drop[p] < 0.1f);
    const bool els    = (!low) && (r_else[p] < 0.3f);

    int cd = strong ? (aug_choice[p] + 1) : 0;   // 1..4
    if (els)  cd = slight_choice[p] + 5;         // 5..6
    if (drop) cd = 7;
    code[p] = cd;
    mpp[p]  = m;
}

// ---------------- Kernel 3: apply augmentation, one block per (b, patch) ----------------
__global__ __launch_bounds__(256) void k_apply(const float* __restrict__ img,
                                               const float* __restrict__ noise,
                                               const float* __restrict__ bright_f,
                                               const float* __restrict__ contrast_f,
                                               const float* __restrict__ slight_f,
                                               const int* __restrict__ code,
                                               const float* __restrict__ mpp,
                                               float* __restrict__ out) {
    __shared__ float tile[PSZ * PSZ];
    const int bi = blockIdx.x;
    const int b = bi >> 12;            // / 4096
    const int p = bi & (NP - 1);
    const int ph = p >> 6, pw = p & 63;
    const int t = threadIdx.x;
    const int r = t >> 4, c = t & 15;
    const size_t gbase = (size_t)b * HH * WW + (size_t)ph * PSZ * WW + (size_t)pw * PSZ;
    const size_t g = gbase + (size_t)r * WW + c;

    const int cd = code[p];            // uniform across the block

    if (cd == 7) {                     // dropped patch: no loads at all
        __builtin_nontemporal_store(0.0f, &out[g]);
        return;
    }

    // Stage the 1KB patch into LDS (async DMA path on gfx1250: ASYNCcnt-tracked)
#if USE_ASYNC_LDS128
    if (t < 64) {                                   // 2 waves, 1 b128 per lane
        const int row = t >> 2;                     // 0..15
        const int cg  = (t & 3) << 2;               // 0,4,8,12
        __builtin_amdgcn_global_load_async_to_lds_b128(
            (__attribute__((address_space(1))) v4i*)(img + gbase + (size_t)row * WW + cg),
            (__attribute__((address_space(3))) v4i*)(&tile[t << 2]),
            0, 0);
        WAIT_ASYNC();
    }
#elif USE_ASYNC_LDS
    __builtin_amdgcn_global_load_async_to_lds_b32(
        (__attribute__((address_space(1))) int*)(img + g),
        (__attribute__((address_space(3))) int*)(&tile[t]),
        0, 0);
    WAIT_ASYNC();
#else
    tile[t] = img[g];
#endif
    __syncthreads();

    const float x = tile[t];
    float o;
    switch (cd) {                       // block-uniform branch
    case 1:
        o = clamp01(x + noise[g] * 0.1f);
        break;
    case 2: {                           // 3x3 zero-padded mean confined to the patch
        float s = 0.0f;
#pragma unroll
        for (int dr = -1; dr <= 1; ++dr) {
#pragma unroll
            for (int dc = -1; dc <= 1; ++dc) {
                const int rr = r + dr, cc = c + dc;
                if (rr >= 0 && rr < PSZ && cc >= 0 && cc < PSZ)
                    s += tile[rr * PSZ + cc];
            }
        }
        o = s * (1.0f / 9.0f);
        break;
    }
    case 3:
        o = clamp01(x * bright_f[p]);
        break;
    case 4: {
        const float m = mpp[p];
        o = clamp01((x - m) * contrast_f[p] + m);
        break;
    }
    case 5:
        o = clamp01(x + noise[g] * 0.05f);
        break;
    case 6:
        o = clamp01(x * slight_f[p]);
        break;
    default:
        o = x;
        break;
    }
    // streaming store: don't evict img from the 192MB L2
    __builtin_nontemporal_store(o, &out[g]);
}

extern "C" void kernel_launch(void* const* d_in, const int* in_sizes, int n_in,
                              void* d_out, int out_size, void* d_ws, size_t ws_size,
                              hipStream_t stream) {
    (void)in_sizes; (void)n_in; (void)out_size; (void)ws_size;
    const float* img        = (const float*)d_in[0];
    const float* noise      = (const float*)d_in[1];
    const float* r_strong   = (const float*)d_in[2];
    const float* r_drop     = (const float*)d_in[3];
    const float* r_else     = (const float*)d_in[4];
    const float* bright_f   = (const float*)d_in[5];
    const float* contrast_f = (const float*)d_in[6];
    const float* slight_f   = (const float*)d_in[7];
    const int*   aug_choice = (const int*)d_in[8];
    const int*   slight_ch  = (const int*)d_in[9];
    float* out = (float*)d_out;

    // workspace layout: [qual_b NBP][mean_b NBP][code NP][mpp NP]  (~1.03 MB)
    float* ws_q    = (float*)d_ws;
    float* ws_m    = ws_q + NBP;
    int*   ws_code = (int*)(ws_m + NBP);
    float* ws_mpp  = (float*)(ws_code + NP);

    k_patch_stats<<<NBP / 8, 256, 0, stream>>>(img, ws_q, ws_m);
    k_code<<<(NP + 255) / 256, 256, 0, stream>>>(ws_q, ws_m, r_strong, r_drop, r_else,
                                                 aug_choice, slight_ch, ws_code, ws_mpp);
    k_apply<<<NBP, 256, 0, stream>>>(img, noise, bright_f, contrast_f, slight_f,
                                     ws_code, ws_mpp, out);
}